// KWinnersTakeAll_86801289052830
// MI455X (gfx1250) — compile-verified
//
#include <hip/hip_runtime.h>
#include <stdint.h>

#define ROWS  8192
#define COLS  8192
#define KACT  410      // ceil(0.05 * 8192)
#define BLOCK 256
#define WAVES (BLOCK / 32)

__device__ __forceinline__ uint32_t f2key(float f) {
    // order-preserving monotonic map: larger float -> larger unsigned key
    uint32_t u = __float_as_uint(f);
    return (u & 0x80000000u) ? ~u : (u | 0x80000000u);
}
__device__ __forceinline__ float key2f(uint32_t k) {
    uint32_t u = (k & 0x80000000u) ? (k & 0x7FFFFFFFu) : ~k;
    return __uint_as_float(u);
}

__global__ __launch_bounds__(BLOCK)
void kwta_row_kernel(const float* __restrict__ x, float* __restrict__ out) {
    __shared__ __align__(16) float srow[COLS];       // 32 KB row staging
    __shared__ uint32_t histW[WAVES][256];           // 8 KB per-wave histograms
    __shared__ uint32_t hist[256];                   // reduced histogram
    __shared__ uint32_t s_prefix, s_rank;
    __shared__ uint32_t s_cntG, s_cntE, s_max;

    const int tid = threadIdx.x;
    const int wid = tid >> 5;
    const int row = blockIdx.x;
    const float* rowp = x + (size_t)row * COLS;

    // ---- Phase 1: async global -> LDS copy (CDNA5 path, ASYNCcnt) ----
    // GVS form: 64-bit scalar base + 32-bit per-lane offset; 8 b128/thread.
    {
        uint32_t la   = (uint32_t)(uintptr_t)(&srow[0]) + (uint32_t)tid * 16u;
        uint32_t voff = (uint32_t)tid * 16u;
        const uint64_t gbase = (uint64_t)(uintptr_t)rowp;
#pragma unroll
        for (int j = 0; j < COLS / 4 / BLOCK; ++j) {
            asm volatile("global_load_async_to_lds_b128 %0, %1, %2"
                         :: "v"(la), "v"(voff), "s"(gbase) : "memory");
            la   += BLOCK * 16u;
            voff += BLOCK * 16u;
        }
    }
    asm volatile("s_wait_asynccnt 0" ::: "memory");

    if (tid == 0) {
        s_prefix = 0u;
        s_rank   = (uint32_t)(KACT - 1);   // descending rank 409 -> topk[k-1]
        s_cntG = 0u; s_cntE = 0u; s_max = 0u;
    }
    __syncthreads();

    const float4* sp4 = (const float4*)srow;

    // ---- Phase 2: 4-pass MSB-first radix select (8-bit digits) in LDS ----
    uint32_t mask = 0u;
    for (int shift = 24; shift >= 0; shift -= 8) {
#pragma unroll
        for (int w = 0; w < WAVES; ++w) histW[w][tid] = 0u;
        __syncthreads();

        const uint32_t pfx = s_prefix;
        for (int i = tid; i < COLS / 4; i += BLOCK) {
            float4 v = sp4[i];
            uint32_t k;
            k = f2key(v.x); if ((k & mask) == pfx) atomicAdd(&histW[wid][(k >> shift) & 0xFFu], 1u);
            k = f2key(v.y); if ((k & mask) == pfx) atomicAdd(&histW[wid][(k >> shift) & 0xFFu], 1u);
            k = f2key(v.z); if ((k & mask) == pfx) atomicAdd(&histW[wid][(k >> shift) & 0xFFu], 1u);
            k = f2key(v.w); if ((k & mask) == pfx) atomicAdd(&histW[wid][(k >> shift) & 0xFFu], 1u);
        }
        __syncthreads();

        // reduce per-wave histograms: hist[tid] = sum over waves
        uint32_t h = 0u;
#pragma unroll
        for (int w = 0; w < WAVES; ++w) h += histW[w][tid];
        hist[tid] = h;
        __syncthreads();

        // parallel digit pick: exclusive suffix count for this thread's bin
        uint32_t excl = 0u;
        for (int b = tid + 1; b < 256; ++b) excl += hist[b];  // broadcast reads
        const uint32_t r = s_rank;
        const bool sel = (excl <= r) && (excl + h > r);       // unique winner
        __syncthreads();
        if (sel) {
            s_prefix = pfx | ((uint32_t)tid << shift);
            s_rank   = r - excl;
        }
        mask |= 0xFFu << shift;
        __syncthreads();
    }

    const uint32_t key1 = s_prefix;   // key of topk[k-1] (410th largest)

    // ---- Phase 3: counts + max-below in a single LDS pass ----
    {
        uint32_t g = 0u, e = 0u, lm = 0u;
        for (int i = tid; i < COLS / 4; i += BLOCK) {
            float4 v = sp4[i];
            uint32_t k;
            k = f2key(v.x); g += (k > key1); e += (k == key1); if (k < key1 && k > lm) lm = k;
            k = f2key(v.y); g += (k > key1); e += (k == key1); if (k < key1 && k > lm) lm = k;
            k = f2key(v.z); g += (k > key1); e += (k == key1); if (k < key1 && k > lm) lm = k;
            k = f2key(v.w); g += (k > key1); e += (k == key1); if (k < key1 && k > lm) lm = k;
        }
        atomicAdd(&s_cntG, g);
        atomicAdd(&s_cntE, e);
        atomicMax(&s_max, lm);
    }
    __syncthreads();

    // topk[k] (411th largest): duplicate of key1 if >=411 elements >= key1,
    // otherwise the largest element strictly below key1.
    const uint32_t key2 = (s_cntG + s_cntE > (uint32_t)KACT) ? key1 : s_max;
    const float thr = 0.5f * (key2f(key1) + key2f(key2));

    // ---- Phase 4: mask output, 128-bit stores straight from LDS ----
    float4* __restrict__ op = (float4*)(out + (size_t)row * COLS);
    for (int i = tid; i < COLS / 4; i += BLOCK) {
        float4 v = sp4[i];
        float4 o;
        o.x = (v.x > thr) ? 1.0f : 0.0f;
        o.y = (v.y > thr) ? 1.0f : 0.0f;
        o.z = (v.z > thr) ? 1.0f : 0.0f;
        o.w = (v.w > thr) ? 1.0f : 0.0f;
        op[i] = o;
    }
}

extern "C" void kernel_launch(void* const* d_in, const int* in_sizes, int n_in,
                              void* d_out, int out_size, void* d_ws, size_t ws_size,
                              hipStream_t stream) {
    (void)in_sizes; (void)n_in; (void)out_size; (void)d_ws; (void)ws_size;
    const float* x = (const float*)d_in[0];
    float* out = (float*)d_out;
    kwta_row_kernel<<<ROWS, BLOCK, 0, stream>>>(x, out);
}